// EdgeScoreGNN_28810640622035
// MI455X (gfx1250) — compile-verified
//
#include <hip/hip_runtime.h>
#include <hip/hip_bf16.h>

typedef __attribute__((ext_vector_type(2))) float v2f;
typedef __attribute__((ext_vector_type(8))) float v8f;

#define IN_DIM 128
#define HID    32

// ---------------- zero fill (deg/acc1/acc2 region) ----------------
__global__ void k_zero_u32(unsigned* __restrict__ p, long long nelem) {
  long long i = blockIdx.x * (long long)blockDim.x + threadIdx.x;
  if (i < nelem) p[i] = 0u;
}

// ---------------- h = x @ W1 via V_WMMA_F32_16X16X4_F32 ----------------
// One wave computes a 16x32 tile of h (two 16x16 accumulators).
// A (16x4 f32): lane = hi*16+lr holds A[lr][2*hi], A[lr][2*hi+1]
// B (4x16 f32): VGPR0 = row k0+2*hi, VGPR1 = row k0+2*hi+1, column lr
// C (16x16 f32): VGPR r = row r (lanes 0-15) / row r+8 (lanes 16-31)
__global__ void k_gemm_wmma(const float* __restrict__ x, const float* __restrict__ W,
                            float* __restrict__ h, int n) {
  int gwave = (int)((blockIdx.x * (long long)blockDim.x + threadIdx.x) >> 5);
  int lane  = threadIdx.x & 31;
  int m0 = gwave * 16;
  if (m0 >= n) return;                 // wave-uniform: EXEC stays all-1s below
  int lr = lane & 15;
  int hi = lane >> 4;

  const float* arow  = x + (size_t)(m0 + lr) * IN_DIM + 2 * hi;
  const float* bbase = W + (size_t)(2 * hi) * HID + lr;

  v8f c0 = {0.f,0.f,0.f,0.f,0.f,0.f,0.f,0.f};
  v8f c1 = {0.f,0.f,0.f,0.f,0.f,0.f,0.f,0.f};

#pragma unroll
  for (int k0 = 0; k0 < IN_DIM; k0 += 4) {
    v2f a;  a[0]  = arow[k0];            a[1]  = arow[k0 + 1];
    const float* bp = bbase + (size_t)k0 * HID;
    v2f b0; b0[0] = bp[0];               b0[1] = bp[HID];
    v2f b1; b1[0] = bp[16];              b1[1] = bp[HID + 16];
    c0 = __builtin_amdgcn_wmma_f32_16x16x4_f32(false, a, false, b0, (short)0, c0, false, false);
    c1 = __builtin_amdgcn_wmma_f32_16x16x4_f32(false, a, false, b1, (short)0, c1, false, false);
  }

  float* hbase = h + (size_t)m0 * HID;
#pragma unroll
  for (int r = 0; r < 8; ++r) {
    int row = r + 8 * hi;
    hbase[(size_t)row * HID + lr]      = c0[r];
    hbase[(size_t)row * HID + 16 + lr] = c1[r];
  }
}

// ---------------- degree on dst (self-loop added analytically later) ----------------
__global__ void k_deg(const long long* __restrict__ dst, unsigned* __restrict__ deg, long long E) {
  long long e = blockIdx.x * (long long)blockDim.x + threadIdx.x;
  if (e < E) atomicAdd(&deg[(int)dst[e]], 1u);
}

__global__ void k_dinv(const unsigned* __restrict__ deg, float* __restrict__ dinv, int n) {
  int i = blockIdx.x * blockDim.x + threadIdx.x;
  if (i < n) dinv[i] = rsqrtf((float)(deg[i] + 1u));   // +1 = self-loop; deg>0 always
}

// ---------------- layer-1 propagation: one 32-lane group per edge ----------------
__global__ void k_prop1(const long long* __restrict__ src, const long long* __restrict__ dst,
                        const float* __restrict__ dinv, const float* __restrict__ h,
                        float* __restrict__ acc1, long long E) {
  long long t = blockIdx.x * (long long)blockDim.x + threadIdx.x;
  long long e = t >> 5;
  if (e >= E) return;
  int j = (int)(t & 31);
  int s = (int)src[e];
  int d = (int)dst[e];
  float w = dinv[s] * dinv[d];
  atomicAdd(&acc1[(size_t)d * HID + j], h[(size_t)s * HID + j] * w);
}

// ---------------- layer-1 epilogue fused with layer-2 GEMV: one wave per node ----------------
__global__ void k_finish1(const float* __restrict__ acc1, const float* __restrict__ hmat,
                          const float* __restrict__ dinv, const float* __restrict__ b1,
                          const float* __restrict__ W2, float* __restrict__ g, int n) {
  int lane = threadIdx.x & 31;
  int i = blockIdx.x * (blockDim.x >> 5) + (threadIdx.x >> 5);
  if (i >= n) return;
  float di = dinv[i];
  size_t off = (size_t)i * HID + lane;
  float v = acc1[off] + hmat[off] * di * di + b1[lane];  // scatter-sum + self-loop + bias
  v = fmaxf(v, 0.0f);                                     // ReLU
  float t = v * W2[lane];                                 // h2 @ W2 (wave reduction)
  for (int o = 16; o > 0; o >>= 1) t += __shfl_down(t, o, 32);
  if (lane == 0) g[i] = t;
}

// ---------------- layer-2 propagation: one thread per edge (scalar feature) ----------------
__global__ void k_prop2(const long long* __restrict__ src, const long long* __restrict__ dst,
                        const float* __restrict__ dinv, const float* __restrict__ g,
                        float* __restrict__ acc2, long long E) {
  long long e = blockIdx.x * (long long)blockDim.x + threadIdx.x;
  if (e >= E) return;
  int s = (int)src[e];
  int d = (int)dst[e];
  atomicAdd(&acc2[d], g[s] * dinv[s] * dinv[d]);
}

// ---------------- sigmoid epilogue ----------------
__global__ void k_final(const float* __restrict__ acc2, const float* __restrict__ g,
                        const float* __restrict__ dinv, const float* __restrict__ b2,
                        float* __restrict__ out, int n) {
  int i = blockIdx.x * blockDim.x + threadIdx.x;
  if (i >= n) return;
  float di = dinv[i];
  float z = acc2[i] + g[i] * di * di + b2[0];
  out[i] = 1.0f / (1.0f + __expf(-z));
}

extern "C" void kernel_launch(void* const* d_in, const int* in_sizes, int n_in,
                              void* d_out, int out_size, void* d_ws, size_t ws_size,
                              hipStream_t stream) {
  const float*     x   = (const float*)d_in[0];
  const long long* ei  = (const long long*)d_in[1];   // int64 [2, E]
  const float*     W1  = (const float*)d_in[2];
  const float*     b1  = (const float*)d_in[3];
  const float*     W2  = (const float*)d_in[4];
  const float*     b2  = (const float*)d_in[5];
  float*           out = (float*)d_out;

  const int       n   = in_sizes[0] / IN_DIM;   // 100000 nodes
  const long long E   = in_sizes[1] / 2;        // 3.2M edges
  const long long* src = ei;
  const long long* dst = ei + E;

  // Workspace layout (floats): [h 32n][deg n][acc1 32n][acc2 n][dinv n][g n] = 68n * 4B ~ 27 MB
  float*    ws   = (float*)d_ws;
  float*    hbuf = ws;
  unsigned* deg  = (unsigned*)(ws + (size_t)32 * n);
  float*    acc1 = ws + (size_t)33 * n;
  float*    acc2 = ws + (size_t)65 * n;
  float*    dinv = ws + (size_t)66 * n;
  float*    g    = ws + (size_t)67 * n;

  // zero deg + acc1 + acc2 (contiguous 34n words)
  long long nz = (long long)34 * n;
  k_zero_u32<<<(unsigned)((nz + 255) / 256), 256, 0, stream>>>(deg, nz);

  // dense GEMM on the WMMA pipe: ceil(n/16) waves, 8 waves (256 threads) per block
  int waves = (n + 15) / 16;
  k_gemm_wmma<<<(waves + 7) / 8, 256, 0, stream>>>(x, W1, hbuf, n);

  k_deg<<<(unsigned)((E + 255) / 256), 256, 0, stream>>>(dst, deg, E);
  k_dinv<<<(n + 255) / 256, 256, 0, stream>>>(deg, dinv, n);

  long long t1 = E * 32;   // (edge, feature) pairs
  k_prop1<<<(unsigned)((t1 + 255) / 256), 256, 0, stream>>>(src, dst, dinv, hbuf, acc1, E);

  k_finish1<<<(n + 7) / 8, 256, 0, stream>>>(acc1, hbuf, dinv, b1, W2, g, n);

  k_prop2<<<(unsigned)((E + 255) / 256), 256, 0, stream>>>(src, dst, dinv, g, acc2, E);

  k_final<<<(n + 255) / 256, 256, 0, stream>>>(acc2, g, dinv, b2, out, n);
}